// GaussianTruncRenormReadPort_28243704938564
// MI455X (gfx1250) — compile-verified
//
#include <hip/hip_runtime.h>
#include <stdint.h>

// GaussianTruncRenormReadPort on MI455X (gfx1250).
// L2-gather-bound (822 MB gathered reads vs 0.41 GFLOP; 51.5 MB table fits L2).
// Path: async global->LDS streaming (ASYNCcnt) + wave32 per-row reduction.

#define SIGMA_F   8.0f
#define RADIUS    24
#define WIN       49          // 2*RADIUS + 1
#define DIM       256
#define WPB       8           // waves per block (256 threads)
#define NBUF      4           // LDS ring depth (rows in flight: 3)

// 16-byte vector type matching the builtin's expected pointee.
typedef int v4i_t __attribute__((vector_size(16)));
typedef __attribute__((address_space(1))) v4i_t g_v4i;
typedef __attribute__((address_space(3))) v4i_t l_v4i;
typedef __attribute__((address_space(3))) void lds_void;

// ---- CDNA5 async copy: 16B per lane, global -> LDS, tracked by ASYNCcnt ----
__device__ __forceinline__ void async_copy_b128(void* lds_ptr, const void* gptr) {
#if __has_builtin(__builtin_amdgcn_global_load_async_to_lds_b128)
    __builtin_amdgcn_global_load_async_to_lds_b128(
        (g_v4i*)gptr, (l_v4i*)lds_ptr, 0, 0);
#else
    unsigned lds_off = (unsigned)(uintptr_t)(lds_void*)lds_ptr;
    uint64_t gaddr   = (uint64_t)(uintptr_t)gptr;
    asm volatile("global_load_async_to_lds_b128 %0, %1, off"
                 :: "v"(lds_off), "v"(gaddr) : "memory");
#endif
}

#if __has_builtin(__builtin_amdgcn_s_wait_asynccnt)
#define WAIT_ASYNC(n) __builtin_amdgcn_s_wait_asynccnt(n)
#else
#define WAIT_ASYNC(n) asm volatile("s_wait_asynccnt %0" :: "n"(n) : "memory")
#endif

#if __has_builtin(__builtin_amdgcn_s_wait_dscnt)
#define WAIT_DS0() __builtin_amdgcn_s_wait_dscnt(0)
#else
#define WAIT_DS0() asm volatile("s_wait_dscnt 0" ::: "memory")
#endif

__global__ __launch_bounds__(WPB * 32)
void gauss_gather_kernel(const float* __restrict__ s,
                         const int*   __restrict__ tok,
                         const float* __restrict__ emb,
                         float*       __restrict__ out,
                         int B, int L) {
    // Per-wave private ring buffers: 4 x 1KB rows.
    __shared__ __align__(16) float smem[WPB][NBUF][DIM];

    const int lane = threadIdx.x & 31;
    const int wid  = threadIdx.x >> 5;
    const int b    = blockIdx.x * WPB + wid;
    if (b >= B) return;

    const float sv   = s[b];                 // wave-uniform
    const int   base = (int)floorf(sv);
    const float frac = sv - (float)base;     // dist(k) = (k - RADIUS) - frac

    // Preload the 49 token ids across lanes (2 coalesced loads), broadcast later.
    int i0 = base - RADIUS + lane;
    int i1 = i0 + 32;
    int c0 = min(max(i0, 0), L - 1);
    int c1 = min(max(i1, 0), L - 1);
    int t0 = tok[c0];
    int t1 = tok[c1];

    float* mybuf = &smem[wid][0][0];

    // Issue both 16B async copies for window slot k into ring buffer k&3.
    // Row base address is an SGPR (readlane of the token id), lane offset in VGPR.
    auto issue_row = [&](int k) {
        int tk = (k < 32) ? __builtin_amdgcn_readlane(t0, k)
                          : __builtin_amdgcn_readlane(t1, k - 32);
        const char* rp = (const char*)(emb + (size_t)tk * DIM);
        char* lb = (char*)(mybuf + (size_t)(k & (NBUF - 1)) * DIM);
        async_copy_b128(lb + lane * 32,      rp + lane * 32);
        async_copy_b128(lb + lane * 32 + 16, rp + lane * 32 + 16);
    };

    issue_row(0);
    issue_row(1);
    issue_row(2);

    float ax0 = 0.f, ax1 = 0.f, ax2 = 0.f, ax3 = 0.f;
    float ay0 = 0.f, ay1 = 0.f, ay2 = 0.f, ay3 = 0.f;
    float wsum = 0.f;

    const float inv_sigma = 1.0f / SIGMA_F;

#pragma unroll
    for (int k = 0; k < WIN; ++k) {
        // Refill pipeline 3 rows ahead. WAR guard: our previous DS reads of the
        // buffer being overwritten must have retired (async LDS writes are
        // unordered vs this wave's ds_loads).
        if (k + 3 < WIN) {
            WAIT_DS0();
            issue_row(k + 3);
        }
        // Drain down to the copies that belong to rows > k (2 loads per row).
        if      (k < WIN - 3) WAIT_ASYNC(6);
        else if (k == WIN - 3) WAIT_ASYNC(4);
        else if (k == WIN - 2) WAIT_ASYNC(2);
        else                   WAIT_ASYNC(0);

        // Gaussian weight for slot k (uniform across the wave).
        int   idx = base + k - RADIUS;
        float d   = (float)(k - RADIUS) - frac;
        float z   = d * inv_sigma;
        float wk  = __expf(-0.5f * z * z);
        if (idx < 0 || idx >= L) wk = 0.f;
        wsum += wk;

        const float4* lb =
            (const float4*)(mybuf + (size_t)(k & (NBUF - 1)) * DIM + lane * 8);
        float4 v0 = lb[0];
        float4 v1 = lb[1];
        ax0 = fmaf(wk, v0.x, ax0); ax1 = fmaf(wk, v0.y, ax1);
        ax2 = fmaf(wk, v0.z, ax2); ax3 = fmaf(wk, v0.w, ax3);
        ay0 = fmaf(wk, v1.x, ay0); ay1 = fmaf(wk, v1.y, ay1);
        ay2 = fmaf(wk, v1.z, ay2); ay3 = fmaf(wk, v1.w, ay3);
    }

    const float inv = 1.0f / fmaxf(wsum, 1e-8f);
    float4 o0 = make_float4(ax0 * inv, ax1 * inv, ax2 * inv, ax3 * inv);
    float4 o1 = make_float4(ay0 * inv, ay1 * inv, ay2 * inv, ay3 * inv);

    float4* op = (float4*)(out + (size_t)b * DIM + lane * 8);
    op[0] = o0;
    op[1] = o1;
}

extern "C" void kernel_launch(void* const* d_in, const int* in_sizes, int n_in,
                              void* d_out, int out_size, void* d_ws, size_t ws_size,
                              hipStream_t stream) {
    const float* s   = (const float*)d_in[0];   // [B] f32
    const int*   tok = (const int*)  d_in[1];   // [L] i32
    const float* emb = (const float*)d_in[2];   // [VOCAB, 256] f32
    float*       out = (float*)d_out;           // [B, 256] f32

    const int B = in_sizes[0];
    const int L = in_sizes[1];

    dim3 block(WPB * 32);
    dim3 grid((B + WPB - 1) / WPB);
    gauss_gather_kernel<<<grid, block, 0, stream>>>(s, tok, emb, out, B, L);
}